// SecondaryStructureEncoder_24601572671727
// MI455X (gfx1250) — compile-verified
//
#include <hip/hip_runtime.h>

typedef __attribute__((ext_vector_type(2))) float v2f;
typedef __attribute__((ext_vector_type(8))) float v8f;

#define N_NODES 50000
#define N_EDGES 600000
#define D 128

// ---------------- utility: zero a float range (grid-stride) ----------------
__global__ void zero_f32(float* __restrict__ p, long n) {
    long i = (long)blockIdx.x * blockDim.x + threadIdx.x;
    long stride = (long)gridDim.x * blockDim.x;
    for (; i < n; i += stride) p[i] = 0.0f;
}

// ---------------- degree count + finalize ----------------
__global__ void deg_count(const int* __restrict__ row, float* __restrict__ deg, int nE) {
    int e = blockIdx.x * blockDim.x + threadIdx.x;
    if (e < nE) unsafeAtomicAdd(&deg[row[e]], 1.0f);
}

__global__ void deg_finalize(float* __restrict__ deg, int n) {
    int i = blockIdx.x * blockDim.x + threadIdx.x;
    if (i < n) deg[i] = 1.0f / fmaxf(deg[i], 1.0f);
}

// ---------------- scatter-add aggregation: dst[row[e]] += src[col[e]] ----------------
// one wave (32 lanes) per edge; each lane moves 4 consecutive floats (float4 gather,
// 4x global_atomic_add_f32 scatter). 8 edges per 256-thread block.
__global__ __launch_bounds__(256)
void scatter_add(const int* __restrict__ row, const int* __restrict__ col,
                 const float* __restrict__ src, float* __restrict__ dst, int nE) {
    int e    = blockIdx.x * (blockDim.x >> 5) + (threadIdx.x >> 5);
    int lane = threadIdx.x & 31;
    if (e >= nE) return;
    int r = row[e];
    int c = col[e];
    const float4* s = (const float4*)(src + (size_t)c * D);
    float4 v = s[lane];
    float* d = dst + (size_t)r * D + lane * 4;
    unsafeAtomicAdd(d + 0, v.x);
    unsafeAtomicAdd(d + 1, v.y);
    unsafeAtomicAdd(d + 2, v.z);
    unsafeAtomicAdd(d + 3, v.w);
}

// ---------------- fused (agg * dinv) @ W^T + bias, ReLU, via fp32 WMMA ----------------
// One wave per 16-node tile; computes all 8 output 16x16 tiles (A reused 8x).
// V_WMMA_F32_16X16X4_F32 operand layouts (ISA 7.12.2):
//   A 16x4 : lane l -> m = l%16, VGPR v holds A[m][k + 2*(l/16) + v]
//   B 4x16 : lane l -> n = l%16, VGPR v holds B[k + 2*(l/16) + v][n]  (B = W^T -> W[n][k])
//   C/D    : VGPR r, lane l -> M = r + 8*(l/16), N = l%16
__global__ __launch_bounds__(32)
void gemm16_relu(const float* __restrict__ A,     // [N_NODES, D] aggregated features
                 const float* __restrict__ dinv,  // [N_NODES] 1/max(deg,1)
                 const float* __restrict__ W,     // [D, D] row-major (out = Anorm @ W^T)
                 const float* __restrict__ bias,  // [D]
                 float* __restrict__ out)         // [N_NODES, D]
{
    const int mt   = blockIdx.x;        // node tile 0..3124
    const int lane = threadIdx.x;       // 0..31
    const int lm   = lane & 15;         // m for A, n for B, N for C/D
    const int kh   = lane >> 4;         // K half select

    const int mrow = mt * 16 + lm;
    const float di = dinv[mrow];
    const float* arow = A + (size_t)mrow * D;

    v8f acc[8];
#pragma unroll
    for (int nt = 0; nt < 8; ++nt) {
        float bv = bias[nt * 16 + lm];  // bias depends only on N
#pragma unroll
        for (int r = 0; r < 8; ++r) acc[nt][r] = bv;
    }

    for (int k = 0; k < D; k += 4) {
        const int ka = k + 2 * kh;
        v2f a;
        a.x = arow[ka] * di;            // fold 1/deg into A rows
        a.y = arow[ka + 1] * di;
#pragma unroll
        for (int nt = 0; nt < 8; ++nt) {
            const float* wrow = W + (size_t)(nt * 16 + lm) * D + ka;
            v2f b;
            b.x = wrow[0];
            b.y = wrow[1];
            acc[nt] = __builtin_amdgcn_wmma_f32_16x16x4_f32(
                /*neg_a=*/false, a, /*neg_b=*/false, b,
                /*c_mod=*/(short)0, acc[nt], /*reuse_a=*/false, /*reuse_b=*/false);
        }
    }

#pragma unroll
    for (int nt = 0; nt < 8; ++nt) {
#pragma unroll
        for (int r = 0; r < 8; ++r) {
            float v = fmaxf(acc[nt][r], 0.0f);       // ReLU
            int M = r + 8 * kh;
            out[(size_t)(mt * 16 + M) * D + nt * 16 + lm] = v;
        }
    }
}

extern "C" void kernel_launch(void* const* d_in, const int* in_sizes, int n_in,
                              void* d_out, int out_size, void* d_ws, size_t ws_size,
                              hipStream_t stream) {
    const float* x  = (const float*)d_in[0];
    const int*   ei = (const int*)d_in[1];   // [2, N_EDGES] flat
    const float* W1 = (const float*)d_in[2];
    const float* b1 = (const float*)d_in[3];
    const float* W2 = (const float*)d_in[4];
    const float* b2 = (const float*)d_in[5];
    float* out = (float*)d_out;

    const int* row = ei;
    const int* col = ei + N_EDGES;

    // workspace layout: [deg | agg | agg2]; h is staged in d_out (overwritten by layer 2)
    float* deg  = (float*)d_ws;
    float* agg  = deg + N_NODES;
    float* agg2 = agg + (size_t)N_NODES * D;
    float* h    = out;

    const long n_zero = (long)N_NODES + 2L * N_NODES * D;  // deg + agg + agg2
    zero_f32<<<2048, 256, 0, stream>>>(deg, n_zero);

    deg_count<<<(N_EDGES + 255) / 256, 256, 0, stream>>>(row, deg, N_EDGES);
    deg_finalize<<<(N_NODES + 255) / 256, 256, 0, stream>>>(deg, N_NODES);

    // layer 1
    scatter_add<<<N_EDGES / 8, 256, 0, stream>>>(row, col, x, agg, N_EDGES);
    gemm16_relu<<<N_NODES / 16, 32, 0, stream>>>(agg, deg, W1, b1, h);

    // layer 2
    scatter_add<<<N_EDGES / 8, 256, 0, stream>>>(row, col, h, agg2, N_EDGES);
    gemm16_relu<<<N_NODES / 16, 32, 0, stream>>>(agg2, deg, W2, b2, out);
}